// reg_loss_10393820856766
// MI455X (gfx1250) — compile-verified
//
#include <hip/hip_runtime.h>
#include <math.h>

// Problem constants (from reference): B=32, NDEL=16 -> 512 lines, K=10 biquads,
// T=16384 samples, N0=T/8=2048.
#define LINES 512
#define KSEC  10
#define TLEN  16384
#define HN0   2048
#define CH    64          // chunk length (time samples per chunk)
#define NCH   256         // TLEN / CH chunks per line
#define CSTR  65          // padded LDS stride per chunk (65 floats -> conflict-free)
#define NTHR  256         // 8 wave32 per block

typedef float v2f __attribute__((ext_vector_type(2)));
typedef float v8f __attribute__((ext_vector_type(8)));

__device__ inline void mat2_sq(const float m[4], float r[4]) {
    r[0] = m[0]*m[0] + m[1]*m[2];
    r[1] = m[0]*m[1] + m[1]*m[3];
    r[2] = m[2]*m[0] + m[3]*m[2];
    r[3] = m[2]*m[1] + m[3]*m[3];
}

// One workgroup per filter line. Signal stays in LDS for all 10 sections.
// Per section:
//   Phase A: per-chunk zero-state response d_c (2-vector), 256 threads parallel.
//   Phase B: matrix-weighted Hillis-Steele scan: s_in(c) = sum_j (A^64)^(c-1-j) d_j.
//   Phase C: Y_chunk = LowerTriToeplitz(h) * X_chunk + Z * s_in  via f32 WMMA 16x16x4,
//            written back in place (each wave owns disjoint chunks).
__global__ __launch_bounds__(NTHR)
void iir_cascade_kernel(const float* __restrict__ sos, float* __restrict__ gamma_out)
{
    __shared__ float xs[NCH * CSTR];     // padded signal, 66560 B
    __shared__ float hh[KSEC][CH];       // section impulse response h[0..63]
    __shared__ float zx[KSEC][CH];       // Z[t] = c * A^t  (zero-input output rows)
    __shared__ float zy[KSEC][CH];
    __shared__ float sP[KSEC][8][4];     // (A^64)^(2^k), k=0..7 (2x2 row-major)
    __shared__ float cf[KSEC][4];        // a1, a2, g1, g2
    __shared__ float dvx[NCH];           // chunk-state / scan arrays
    __shared__ float dvy[NCH];

    const int tid  = threadIdx.x;
    const int line = blockIdx.x;

    // ---- unit impulse ----
    for (int i = tid; i < NCH * CSTR; i += NTHR) xs[i] = 0.0f;
    if (tid == 0) xs[0] = 1.0f;

    // ---- per-section tables: thread s handles section s ----
    if (tid < KSEC) {
        const int s    = tid;
        const int base = (line * KSEC + s) * 6;
        const float a0i = 1.0f / sos[base + 3];
        const float b0 = sos[base + 0] * a0i;
        const float b1 = sos[base + 1] * a0i;
        const float b2 = sos[base + 2] * a0i;
        const float a1 = sos[base + 4] * a0i;
        const float a2 = sos[base + 5] * a0i;
        // TDF2 state-space: s_t = A s_{t-1} + g x_t ; y_t = [1 0] s_{t-1} + b0 x_t
        // A = [[-a1, 1], [-a2, 0]], g = [b1 - a1 b0, b2 - a2 b0]
        const float g1 = b1 - a1 * b0;
        const float g2 = b2 - a2 * b0;
        cf[s][0] = a1; cf[s][1] = a2; cf[s][2] = g1; cf[s][3] = g2;
        // Z[t] = c A^t (row-vector recurrence), h[0]=b0, h[m]=Z[m-1]·g
        float p = 1.0f, q = 0.0f;
        hh[s][0] = b0;
        for (int t = 0; t < CH; ++t) {
            zx[s][t] = p; zy[s][t] = q;
            if (t < CH - 1) hh[s][t + 1] = p * g1 + q * g2;
            const float pn = -a1 * p - a2 * q;   // [p q]A = [-a1 p - a2 q, p]
            q = p; p = pn;
        }
        // A^64 via 6 squarings, then P_k = (A^64)^(2^k)
        float M[4] = { -a1, 1.0f, -a2, 0.0f };
        float R[4];
        for (int i = 0; i < 6; ++i) {
            mat2_sq(M, R);
            M[0]=R[0]; M[1]=R[1]; M[2]=R[2]; M[3]=R[3];
        }
        sP[s][0][0]=M[0]; sP[s][0][1]=M[1]; sP[s][0][2]=M[2]; sP[s][0][3]=M[3];
        for (int k = 1; k < 8; ++k) {
            mat2_sq(M, R);
            M[0]=R[0]; M[1]=R[1]; M[2]=R[2]; M[3]=R[3];
            sP[s][k][0]=M[0]; sP[s][k][1]=M[1]; sP[s][k][2]=M[2]; sP[s][k][3]=M[3];
        }
    }
    __syncthreads();

    const int lane = tid & 31;
    const int wv   = tid >> 5;
    const int p16  = lane & 15;
    const int half = lane >> 4;

    for (int s = 0; s < KSEC; ++s) {
        const float a1 = cf[s][0], a2 = cf[s][1], g1 = cf[s][2], g2 = cf[s][3];

        // ---- Phase A: zero-state chunk response (thread == chunk) ----
        {
            const float* xc = &xs[tid * CSTR];
            float s1 = 0.0f, s2 = 0.0f;
            #pragma unroll 8
            for (int j = 0; j < CH; ++j) {
                const float x  = xc[j];
                const float n1 = g1 * x - a1 * s1 + s2;
                const float n2 = g2 * x - a2 * s1;
                s1 = n1; s2 = n2;
            }
            dvx[tid] = s1; dvy[tid] = s2;
        }
        __syncthreads();

        // ---- Phase B: matrix-weighted inclusive scan over 256 chunks ----
        for (int k = 0; k < 8; ++k) {
            const int off = 1 << k;
            const bool act = (tid >= off);
            float px = 0.0f, py = 0.0f;
            if (act) { px = dvx[tid - off]; py = dvy[tid - off]; }
            __syncthreads();
            if (act) {
                dvx[tid] += sP[s][k][0]*px + sP[s][k][1]*py;
                dvy[tid] += sP[s][k][2]*px + sP[s][k][3]*py;
            }
            __syncthreads();
        }
        // s_in(chunk c) = dv[c-1]  (dv[-1] = 0)

        // ---- Phase C: Y = T*X + Z*s_in per 16-chunk tile, in place, via WMMA ----
        const float* hs  = hh[s];
        const float* zxs = zx[s];
        const float* zys = zy[s];
        for (int mt = wv * 2; mt < wv * 2 + 2; ++mt) {
            const int m0     = mt * 16;
            const int mchunk = m0 + p16;                     // this lane's D/B column
            const float sxv = (mchunk > 0) ? dvx[mchunk - 1] : 0.0f;
            const float syv = (mchunk > 0) ? dvy[mchunk - 1] : 0.0f;
            const float* xcol = &xs[mchunk * CSTR];

            v8f acc[4];
            #pragma unroll
            for (int nt = 0; nt < 4; ++nt) {
                v8f c = {};
                const int t = nt * 16 + p16;                 // A-matrix row (output time)
                const int kmax = 4 * nt + 3;                 // lower-triangular: j <= t
                for (int k = 0; k <= kmax; ++k) {
                    const int j0 = 4 * k + 2 * half;         // A/B K index for v=0
                    v2f a, b;
                    const int i0 = t - j0;
                    const int i1 = i0 - 1;
                    const int c0 = (i0 >= 0) ? i0 : 0;
                    const int c1 = (i1 >= 0) ? i1 : 0;
                    const float h0 = hs[c0];
                    const float h1 = hs[c1];
                    a[0] = (i0 >= 0) ? h0 : 0.0f;            // A[t][j0]   = h[t-j0]
                    a[1] = (i1 >= 0) ? h1 : 0.0f;            // A[t][j0+1] = h[t-j0-1]
                    b[0] = xcol[j0];                         // B[j0][m]   = x_chunk[j0]
                    b[1] = xcol[j0 + 1];
                    c = __builtin_amdgcn_wmma_f32_16x16x4_f32(
                            false, a, false, b, (short)0, c, false, false);
                }
                // zero-input-response as one extra K-tile (virtual K rows 64,65)
                {
                    v2f a, b;
                    a[0] = half ? 0.0f : zxs[t];
                    a[1] = half ? 0.0f : zys[t];
                    b[0] = half ? 0.0f : sxv;
                    b[1] = half ? 0.0f : syv;
                    c = __builtin_amdgcn_wmma_f32_16x16x4_f32(
                            false, a, false, b, (short)0, c, false, false);
                }
                acc[nt] = c;
            }
            // write back after all reads of this tile's chunks (in-place safe)
            #pragma unroll
            for (int nt = 0; nt < 4; ++nt) {
                float* ycol = &xs[mchunk * CSTR + nt * 16 + 8 * half];
                #pragma unroll
                for (int v = 0; v < 8; ++v) ycol[v] = acc[nt][v];  // D row = v + 8*half
            }
        }
        __syncthreads();
    }

    // ---- gamma = sum|y[T-N0:]| / sum|y[:N0]| ----
    float hsum = 0.0f, tsum = 0.0f;
    for (int t = tid; t < HN0; t += NTHR)
        hsum += fabsf(xs[(t >> 6) * CSTR + (t & 63)]);
    for (int t = TLEN - HN0 + tid; t < TLEN; t += NTHR)
        tsum += fabsf(xs[(t >> 6) * CSTR + (t & 63)]);
    dvx[tid] = hsum; dvy[tid] = tsum;
    __syncthreads();
    for (int o = NTHR / 2; o > 0; o >>= 1) {
        if (tid < o) { dvx[tid] += dvx[tid + o]; dvy[tid] += dvy[tid + o]; }
        __syncthreads();
    }
    if (tid == 0) gamma_out[line] = dvy[0] / dvx[0];
}

// loss = sum_b ( sum_n gamma*exp(gamma) / sum_n exp(gamma) )
__global__ __launch_bounds__(LINES)
void loss_kernel(const float* __restrict__ gamma, float* __restrict__ out)
{
    __shared__ float sgw[LINES];
    __shared__ float sgg[LINES];
    __shared__ float rl[32];
    const int tid = threadIdx.x;
    const float g = gamma[tid];
    const float w = expf(g);
    sgw[tid] = w;
    sgg[tid] = g * w;
    __syncthreads();
    if ((tid & 15) == 0) {
        float sw = 0.0f, sg = 0.0f;
        for (int i = 0; i < 16; ++i) { sw += sgw[tid + i]; sg += sgg[tid + i]; }
        rl[tid >> 4] = sg / sw;
    }
    __syncthreads();
    if (tid == 0) {
        float sum = 0.0f;
        for (int i = 0; i < 32; ++i) sum += rl[i];
        out[0] = sum;
    }
}

extern "C" void kernel_launch(void* const* d_in, const int* in_sizes, int n_in,
                              void* d_out, int out_size, void* d_ws, size_t ws_size,
                              hipStream_t stream) {
    (void)in_sizes; (void)n_in; (void)out_size; (void)ws_size;
    const float* sos = (const float*)d_in[0];   // [B, NDEL, K, 6] f32
    float* out       = (float*)d_out;           // single f32
    float* gamma_ws  = (float*)d_ws;            // 512 floats scratch

    iir_cascade_kernel<<<LINES, NTHR, 0, stream>>>(sos, gamma_ws);
    loss_kernel<<<1, LINES, 0, stream>>>(gamma_ws, out);
}